// ATSSAssigner_45028437131385
// MI455X (gfx1250) — compile-verified
//
#include <hip/hip_runtime.h>
#include <hip/hip_bf16.h>
#include <math.h>

// ATSS assigner for gfx1250 (MI455X).
// Pipeline:
//  k1: WMMA f32 16x16x4 distance matrix (rank-4 bilinear form) + fused scalar IoU
//  k2: per (b, m, level) top-9 nearest anchors (insertion scan)
//  k2b: threshold = mean + std(ddof=1) over gathered candidate IoUs
//  k3: per (b, anchor) column resolution + all outputs

#define TOPK 9
#define NCLS 80
#define EPSF 1e-9f

typedef __attribute__((ext_vector_type(2))) float v2f;
typedef __attribute__((ext_vector_type(8))) float v8f;

__device__ __forceinline__ float iou_box(float ax0, float ay0, float ax1, float ay1,
                                         float bx0, float by0, float bx1, float by1) {
    float ltx = fmaxf(ax0, bx0), lty = fmaxf(ay0, by0);
    float rbx = fminf(ax1, bx1), rby = fminf(ay1, by1);
    float iw = fmaxf(rbx - ltx, 0.0f), ih = fmaxf(rby - lty, 0.0f);
    float inter = iw * ih;
    float aa = (ax1 - ax0) * (ay1 - ay0);
    float ab = (bx1 - bx0) * (by1 - by0);
    return inter / (aa + ab - inter + EPSF);
}

// ---------------------------------------------------------------------------
// Kernel 1: one wave computes a 16(gt) x 16(anchor) tile.
//   dist^2[m][n] = (-2gx)*ax + (-2gy)*ay + 1*(ax^2+ay^2) + (gx^2+gy^2)*1
// -> V_WMMA_F32_16X16X4_F32 with K=4. IoU computed scalar on the same tile.
// ---------------------------------------------------------------------------
__global__ void k1_dist_iou(const float* __restrict__ anchors,
                            const float* __restrict__ gt,
                            float* __restrict__ ovl,
                            float* __restrict__ dist,
                            int bs, int M, int A, int mtiles, int atiles) {
    int wid  = (blockIdx.x * blockDim.x + threadIdx.x) >> 5;
    int lane = threadIdx.x & 31;
    int total = bs * mtiles * atiles;
    if (wid >= total) return;                 // wave-uniform exit; EXEC stays full
    int at = wid % atiles;
    int mt = (wid / atiles) % mtiles;
    int b  = wid / (atiles * mtiles);

    bool hi = (lane & 16) != 0;
    int  lo = lane & 15;

    int m = mt * 16 + lo; if (m >= M) m = M - 1;
    int n = at * 16 + lo; if (n >= A) n = A - 1;

    const float* g = gt + ((size_t)(b * M + m)) * 4;
    float gx0 = g[0], gy0 = g[1], gx1 = g[2], gy1 = g[3];
    float gcx = 0.5f * (gx0 + gx1), gcy = 0.5f * (gy0 + gy1);

    const float* an = anchors + (size_t)n * 4;
    float ax0 = an[0], ay0 = an[1], ax1 = an[2], ay1 = an[3];
    float acx = 0.5f * (ax0 + ax1), acy = 0.5f * (ay0 + ay1);

    // A-frag (16x4 f32): lanes 0-15 hold K={0,1}, lanes 16-31 hold K={2,3}
    v2f av, bv;
    av.x = hi ? 1.0f                      : -2.0f * gcx;
    av.y = hi ? (gcx * gcx + gcy * gcy)   : -2.0f * gcy;
    // B-frag (4x16 f32), same K split, col = lane&15
    bv.x = hi ? (acx * acx + acy * acy)   : acx;
    bv.y = hi ? 1.0f                      : acy;

    v8f c = {};
    v8f d = __builtin_amdgcn_wmma_f32_16x16x4_f32(
        /*neg_a=*/false, av, /*neg_b=*/false, bv,
        /*c_mod=*/(short)0, c, /*reuse_a=*/false, /*reuse_b=*/false);

    int col = at * 16 + lo;
    if (col < A) {
#pragma unroll
        for (int r = 0; r < 8; ++r) {
            int row = mt * 16 + r + (hi ? 8 : 0);
            if (row < M) {
                size_t off = ((size_t)(b * M + row)) * A + col;
                float d2 = d[r];
                dist[off] = sqrtf(fmaxf(d2, 0.0f));
                const float* gr = gt + ((size_t)(b * M + row)) * 4;
                float hx0 = gr[0], hy0 = gr[1], hx1 = gr[2], hy1 = gr[3];
                ovl[off] = iou_box(hx0, hy0, hx1, hy1, ax0, ay0, ax1, ay1);
            }
        }
    }
}

// ---------------------------------------------------------------------------
// Kernel 2: one thread per (b*M, level): top-9 smallest distances (stable).
// Writes global candidate anchor indices and gathered candidate IoUs.
// ---------------------------------------------------------------------------
__global__ void k2_topk(const float* __restrict__ dist,
                        const float* __restrict__ ovl,
                        const float* __restrict__ mask_gt,
                        const long long* __restrict__ nlevel,
                        int* __restrict__ tidx, float* __restrict__ tg,
                        int bsM, int A, int nlvl) {
    int t = blockIdx.x * blockDim.x + threadIdx.x;
    if (t >= bsM * nlvl) return;
    int lvl = t % nlvl;
    int bm  = t / nlvl;

    int start = 0;
    for (int l = 0; l < lvl; ++l) start += (int)nlevel[l];
    int nb = (int)nlevel[lvl];
    int k  = nb < TOPK ? nb : TOPK;

    float bd[TOPK]; int bi[TOPK];
    for (int j = 0; j < TOPK; ++j) { bd[j] = __builtin_inff(); bi[j] = -1; }

    const float* drow = dist + (size_t)bm * A + start;
    for (int i = 0; i < nb; ++i) {
        float dv = drow[i];
        if (dv < bd[TOPK - 1]) {
            int j = TOPK - 1;
            while (j > 0 && dv < bd[j - 1]) {     // strict < keeps ties stable
                bd[j] = bd[j - 1]; bi[j] = bi[j - 1]; --j;
            }
            bd[j] = dv; bi[j] = i;
        }
    }

    float mk = mask_gt[bm];
    int base = bm * (nlvl * TOPK) + lvl * TOPK;
    for (int j = 0; j < TOPK; ++j) {
        int gi = (j < k && bi[j] >= 0) ? (start + bi[j]) : -1;
        tidx[base + j] = gi;
        tg[base + j]   = (gi >= 0 && mk > 0.0f) ? ovl[(size_t)bm * A + gi] : 0.0f;
    }
}

// ---------------------------------------------------------------------------
// Kernel 2b: threshold per (b, m): mean + std(ddof=1) over valid candidates.
// ---------------------------------------------------------------------------
__global__ void k2b_thr(const int* __restrict__ tidx, const float* __restrict__ tg,
                        float* __restrict__ thr, int bsM, int KT) {
    int t = blockIdx.x * blockDim.x + threadIdx.x;
    if (t >= bsM) return;
    const int* ip = tidx + (size_t)t * KT;
    const float* gp = tg + (size_t)t * KT;
    float s = 0.0f; int n = 0;
    for (int j = 0; j < KT; ++j) if (ip[j] >= 0) { s += gp[j]; ++n; }
    float mean = (n > 0) ? s / (float)n : 0.0f;
    float v = 0.0f;
    for (int j = 0; j < KT; ++j) if (ip[j] >= 0) { float dd = gp[j] - mean; v += dd * dd; }
    float sd = (n > 1) ? sqrtf(v / (float)(n - 1)) : 0.0f;
    thr[t] = mean + sd;
}

// ---------------------------------------------------------------------------
// Kernel 3: one thread per (b, anchor). Resolves the whole column over M and
// writes labels / boxes / scores / fg. Candidate metadata staged in LDS.
// ---------------------------------------------------------------------------
__global__ void k3_assign(const float* __restrict__ anchors,
                          const float* __restrict__ gt,
                          const int* __restrict__ glab,
                          const float* __restrict__ mask_gt,
                          const float* __restrict__ pred,
                          const float* __restrict__ ovl,
                          const float* __restrict__ thr,
                          const int* __restrict__ tidx,
                          float* __restrict__ out_lab, float* __restrict__ out_box,
                          float* __restrict__ out_sc,  float* __restrict__ out_fg,
                          int bs, int M, int A, int KT) {
    extern __shared__ char smem[];
    int*   s_idx  = (int*)smem;               // M*KT
    float* s_thr  = (float*)(s_idx + M * KT); // M
    float* s_mask = s_thr + M;                // M
    int*   s_lab  = (int*)(s_mask + M);       // M
    float* s_box  = (float*)(s_lab + M);      // M*4

    int b = blockIdx.y;
    for (int i = threadIdx.x; i < M * KT; i += blockDim.x)
        s_idx[i] = tidx[(size_t)(b * M) * KT + i];
    for (int i = threadIdx.x; i < M; i += blockDim.x) {
        s_thr[i]  = thr[b * M + i];
        s_mask[i] = mask_gt[b * M + i];
        s_lab[i]  = glab[b * M + i];
        s_box[4 * i + 0] = gt[((size_t)(b * M + i)) * 4 + 0];
        s_box[4 * i + 1] = gt[((size_t)(b * M + i)) * 4 + 1];
        s_box[4 * i + 2] = gt[((size_t)(b * M + i)) * 4 + 2];
        s_box[4 * i + 3] = gt[((size_t)(b * M + i)) * 4 + 3];
    }
    __syncthreads();

    int a = blockIdx.x * blockDim.x + threadIdx.x;
    if (a >= A) return;

    const float* an = anchors + (size_t)a * 4;
    float ax0 = an[0], ay0 = an[1], ax1 = an[2], ay1 = an[3];
    float acx = 0.5f * (ax0 + ax1), acy = 0.5f * (ay0 + ay1);

    int sum = 0, first_pos = -1, amax = 0;
    float maxov = -__builtin_inff();
    for (int m = 0; m < M; ++m) {
        float ov = ovl[((size_t)(b * M + m)) * A + a];
        if (ov > maxov) { maxov = ov; amax = m; }   // strict > : first argmax
        bool member = false;
        for (int j = 0; j < KT; ++j) member |= (s_idx[m * KT + j] == a);
        float bx0 = s_box[4 * m + 0], by0 = s_box[4 * m + 1];
        float bx1 = s_box[4 * m + 2], by1 = s_box[4 * m + 3];
        float mn = fminf(fminf(acx - bx0, acy - by0), fminf(bx1 - acx, by1 - acy));
        bool mp = member && (ov > s_thr[m]) && (mn > EPSF) && (s_mask[m] > 0.0f);
        if (mp) { if (first_pos < 0) first_pos = m; ++sum; }
    }

    int gt_idx, fg;
    if (sum > 1)      { gt_idx = amax;      fg = 1; }   // multi -> max-IoU GT
    else if (sum == 1){ gt_idx = first_pos; fg = 1; }
    else              { gt_idx = 0;         fg = 0; }   // argmax of zeros == 0

    int lab = fg ? s_lab[gt_idx] : NCLS;
    size_t oi = (size_t)b * A + a;
    out_lab[oi] = (float)lab;
    float bx0 = s_box[4 * gt_idx + 0], by0 = s_box[4 * gt_idx + 1];
    float bx1 = s_box[4 * gt_idx + 2], by1 = s_box[4 * gt_idx + 3];
    out_box[oi * 4 + 0] = bx0; out_box[oi * 4 + 1] = by0;
    out_box[oi * 4 + 2] = bx1; out_box[oi * 4 + 3] = by1;

    float iouv = 0.0f;
    if (fg) {
        const float* p = pred + ((size_t)b * A + a) * 4;
        iouv = iou_box(bx0, by0, bx1, by1, p[0], p[1], p[2], p[3]);
    }
    float* sp = out_sc + oi * NCLS;
    for (int c = 0; c < NCLS; ++c)
        sp[c] = (fg && c == lab) ? iouv : 0.0f;
    out_fg[oi] = fg ? 1.0f : 0.0f;
}

// ---------------------------------------------------------------------------
extern "C" void kernel_launch(void* const* d_in, const int* in_sizes, int n_in,
                              void* d_out, int out_size, void* d_ws, size_t ws_size,
                              hipStream_t stream) {
    const float*     anchors = (const float*)d_in[0];
    const long long* nlevel  = (const long long*)d_in[1];
    const int*       glab    = (const int*)d_in[2];
    const float*     gt      = (const float*)d_in[3];
    const float*     mask_gt = (const float*)d_in[4];
    const float*     pred    = (const float*)d_in[5];

    int A    = in_sizes[0] / 4;          // 33600
    int nlvl = in_sizes[1];              // 3
    int bsM  = in_sizes[2];              // bs*M = 512
    int bs   = in_sizes[5] / (4 * A);    // 8
    int M    = bsM / bs;                 // 64
    int KT   = nlvl * TOPK;              // 27

    // workspace layout
    char* ws = (char*)d_ws;
    size_t szD = (size_t)bsM * A * sizeof(float);
    float* ovl  = (float*)ws; ws += szD;
    float* dist = (float*)ws; ws += szD;
    int*   tidx = (int*)ws;   ws += (size_t)bsM * KT * sizeof(int);
    float* tg   = (float*)ws; ws += (size_t)bsM * KT * sizeof(float);
    float* thr  = (float*)ws; ws += (size_t)bsM * sizeof(float);

    // output layout (flat, return order, all as f32)
    float* out     = (float*)d_out;
    float* out_lab = out;
    float* out_box = out_lab + (size_t)bs * A;
    float* out_sc  = out_box + (size_t)bs * A * 4;
    float* out_fg  = out_sc  + (size_t)bs * A * NCLS;

    // k1: one wave per 16x16 tile, 8 waves per 256-thread block
    int mtiles = (M + 15) / 16, atiles = (A + 15) / 16;
    int tiles  = bs * mtiles * atiles;
    int blk1   = (tiles + 7) / 8;
    k1_dist_iou<<<blk1, 256, 0, stream>>>(anchors, gt, ovl, dist,
                                          bs, M, A, mtiles, atiles);

    int tot2 = bsM * nlvl;
    k2_topk<<<(tot2 + 63) / 64, 64, 0, stream>>>(dist, ovl, mask_gt, nlevel,
                                                 tidx, tg, bsM, A, nlvl);

    k2b_thr<<<(bsM + 63) / 64, 64, 0, stream>>>(tidx, tg, thr, bsM, KT);

    dim3 g3((A + 255) / 256, bs);
    size_t sm3 = (size_t)M * KT * sizeof(int) + (size_t)M * 3 * sizeof(float)
               + (size_t)M * 4 * sizeof(float);
    k3_assign<<<g3, 256, sm3, stream>>>(anchors, gt, glab, mask_gt, pred,
                                        ovl, thr, tidx,
                                        out_lab, out_box, out_sc, out_fg,
                                        bs, M, A, KT);
}